// AdditiveAttention_68564857914088
// MI455X (gfx1250) — compile-verified
//
#include <hip/hip_runtime.h>
#include <math.h>
#include <stdint.h>

// ---------------- problem constants (from reference) ----------------
#define LSEQ   512
#define DDIM   256
#define UNITS  256
#define KT     16            // key tile for flash phase (double-buffered)
#define KROW   272           // row stride in floats; 272 % 64 == 16 -> conflict-free
#define NTILE  (LSEQ / KT)

typedef __attribute__((ext_vector_type(2))) float v2f;
typedef __attribute__((ext_vector_type(8))) float v8f;

// ---------------- fast device math ----------------
__device__ __forceinline__ float fast_tanh(float x) {
#if __has_builtin(__builtin_amdgcn_tanhf)
    return __builtin_amdgcn_tanhf(x);      // v_tanh_f32 (CDNA5 TRANS op)
#else
    return tanhf(x);
#endif
}

__device__ __forceinline__ float fast_exp(float x) {
    return __builtin_amdgcn_exp2f(x * 1.4426950408889634f);   // v_exp_f32
}

// ---------------- CDNA5 async global -> LDS (ASYNCcnt path) ----------------
__device__ __forceinline__ void async_load_b128_to_lds(uint32_t lds_byte_addr,
                                                       const void* gaddr) {
    // per-lane: copies 16 bytes from global gaddr to LDS byte address lds_byte_addr, tracked by ASYNCcnt
    asm volatile("global_load_async_to_lds_b128 %0, %1, off"
                 :: "v"(lds_byte_addr), "v"(gaddr)
                 : "memory");
}

__device__ __forceinline__ void wait_async0() {
#if __has_builtin(__builtin_amdgcn_s_wait_asynccnt)
    __builtin_amdgcn_s_wait_asynccnt(0);
#else
    asm volatile("s_wait_asynccnt 0" ::: "memory");
#endif
}

// =====================================================================
// Kernel A: Y = X @ W + bias  via V_WMMA_F32_16X16X4_F32
//   One wave (32 lanes) computes a 16x16 tile of Y, K-loop in steps of 4.
// =====================================================================
__global__ __launch_bounds__(32)
void proj_gemm_wmma(const float* __restrict__ X,
                    const float* __restrict__ W,
                    const float* __restrict__ bias,
                    float* __restrict__ Y)
{
    const int lane = threadIdx.x;
    const int row0 = blockIdx.x * 16;
    const int col0 = blockIdx.y * 16;
    const int hi   = (lane >= 16) ? 1 : 0;
    const int l16  = lane & 15;

    v8f acc = {};
    const float* arow = X + (size_t)(row0 + l16) * UNITS + 2 * hi;
    const float* bcol = W + (size_t)(2 * hi) * UNITS + col0 + l16;

    #pragma unroll 4
    for (int k0 = 0; k0 < UNITS; k0 += 4) {
        v2f a, b;
        a.x = arow[k0 + 0];
        a.y = arow[k0 + 1];
        b.x = bcol[(size_t)(k0 + 0) * UNITS];
        b.y = bcol[(size_t)(k0 + 1) * UNITS];
        acc = __builtin_amdgcn_wmma_f32_16x16x4_f32(
                  false, a, false, b, (short)0, acc, false, false);
    }

    const float bv   = bias[col0 + l16];
    const int rbase  = row0 + (hi ? 8 : 0);
    #pragma unroll
    for (int r = 0; r < 8; ++r)
        Y[(size_t)(rbase + r) * UNITS + col0 + l16] = acc[r] + bv;
}

// =====================================================================
// Kernel B: fused additive-attention row (flash style), double-buffered
// K tiles moved with GLOBAL_LOAD_ASYNC_TO_LDS_B128.
//   One 256-thread workgroup per (b, q). Thread d owns output element d.
//   logits[k] = v_b + sum_u v[u] * tanh(Qp[q,u] + Kp[k,u])
//   online softmax over k, out[d] = sum_k attn[k] * value[b,k,d]
// =====================================================================
__global__ __launch_bounds__(256)
void additive_attn_fused(const float* __restrict__ Qp,   // (B,L,U)
                         const float* __restrict__ Kp,   // (B,L,U)
                         const float* __restrict__ Vv,   // value (B,L,D)
                         const float* __restrict__ vw,   // (U)
                         const float* __restrict__ vb,   // (1)
                         float* __restrict__ out)        // (B,L,D)
{
    __shared__ float q_s[UNITS];
    __shared__ float vw_s[UNITS];
    __shared__ float k_s[2][KT][KROW];
    __shared__ float log_s[KT];

    const int tid = threadIdx.x;
    const int q   = blockIdx.x;
    const int b   = blockIdx.y;
    const int d   = tid;

    q_s[tid]  = Qp[((size_t)b * LSEQ + q) * UNITS + tid];
    vw_s[tid] = vw[tid];
    const float vbias = vb[0];

    const int k_local = tid >> 4;   // 0..15 : which key in the tile
    const int uu      = tid & 15;   // 0..15 : unit-slice within the key

    float m     = -3.0e37f;   // running max
    float denom = 0.0f;
    float acc   = 0.0f;

    const float* kbase = Kp + (size_t)b * LSEQ * UNITS;
    const float* vbase = Vv + (size_t)b * LSEQ * DDIM;

    // per-thread chunk coords for the async tile copy:
    // tile = KT rows x 1024 bytes = 1024 x 16B chunks; 4 chunks per thread
    // chunk c = tid + 256*i -> row = c>>6, 16B-col = c&63
    const uint32_t lds_base = (uint32_t)(uintptr_t)(&k_s[0][0][0]);
    const uint32_t buf_bytes = (uint32_t)(KT * KROW * sizeof(float));

    // ---- prologue: issue async copy of tile 0 into buffer 0 ----
    #pragma unroll
    for (int i = 0; i < 4; ++i) {
        const int c  = tid + 256 * i;
        const int kr = c >> 6;
        const int kc = c & 63;
        async_load_b128_to_lds(lds_base + (uint32_t)(kr * KROW * 4 + kc * 16),
                               kbase + ((size_t)kr * UNITS + (size_t)kc * 4));
    }

    for (int t = 0; t < NTILE; ++t) {
        const int buf = t & 1;
        const int kt  = t * KT;

        wait_async0();        // our async copies done
        __syncthreads();      // everyone's copies visible; prev compute finished

        // ---- issue async copy of next tile into the other buffer ----
        if (t + 1 < NTILE) {
            const uint32_t dst = lds_base + (buf ? 0u : buf_bytes);
            const float*  src = kbase + (size_t)(kt + KT) * UNITS;
            #pragma unroll
            for (int i = 0; i < 4; ++i) {
                const int c  = tid + 256 * i;
                const int kr = c >> 6;
                const int kc = c & 63;
                async_load_b128_to_lds(dst + (uint32_t)(kr * KROW * 4 + kc * 16),
                                       src + ((size_t)kr * UNITS + (size_t)kc * 4));
            }
        }

        // ---- Phase A: logit partials (16 tanh per thread) ----
        float p = 0.0f;
        const float* krow = &k_s[buf][k_local][0];
        #pragma unroll
        for (int j = 0; j < UNITS / 16; ++j) {
            const int u = uu + (j << 4);
            p += vw_s[u] * fast_tanh(q_s[u] + krow[u]);
        }
        // reduce the 16 unit-slices (contiguous lanes within a wave)
        p += __shfl_xor(p, 1, 32);
        p += __shfl_xor(p, 2, 32);
        p += __shfl_xor(p, 4, 32);
        p += __shfl_xor(p, 8, 32);
        if (uu == 0) log_s[k_local] = p + vbias;
        __syncthreads();

        // ---- Phase B: online softmax + value accumulation ----
        float tmax = m;
        #pragma unroll
        for (int kk = 0; kk < KT; ++kk) tmax = fmaxf(tmax, log_s[kk]);

        const float corr = fast_exp(m - tmax);
        denom *= corr;
        acc   *= corr;

        #pragma unroll
        for (int kk = 0; kk < KT; ++kk) {
            const float pe = fast_exp(log_s[kk] - tmax);
            denom += pe;
            acc   += pe * vbase[(size_t)(kt + kk) * DDIM + d];
        }
        m = tmax;
    }

    out[((size_t)b * LSEQ + q) * DDIM + d] = acc / denom;
}

// =====================================================================
extern "C" void kernel_launch(void* const* d_in, const int* in_sizes, int n_in,
                              void* d_out, int out_size, void* d_ws, size_t ws_size,
                              hipStream_t stream)
{
    const float* query = (const float*)d_in[0];   // (B,L,D)
    const float* value = (const float*)d_in[1];   // (B,L,D)
    const float* U_w   = (const float*)d_in[2];   // (D,U)
    const float* U_b   = (const float*)d_in[3];   // (U)
    const float* W_w   = (const float*)d_in[4];   // (D,U)
    const float* W_b   = (const float*)d_in[5];   // (U)
    const float* v_w   = (const float*)d_in[6];   // (U,1)
    const float* v_b   = (const float*)d_in[7];   // (1)

    const int B    = in_sizes[0] / (LSEQ * DDIM);
    const int rows = B * LSEQ;

    float* Qp = (float*)d_ws;                       // rows x UNITS
    float* Kp = Qp + (size_t)rows * UNITS;          // rows x UNITS

    // Projections via WMMA: one wave per 16x16 output tile
    dim3 gProj(rows / 16, UNITS / 16);
    proj_gemm_wmma<<<gProj, 32, 0, stream>>>(query, U_w, U_b, Qp);
    proj_gemm_wmma<<<gProj, 32, 0, stream>>>(value, W_w, W_b, Kp);

    // Fused tanh-score + softmax + weighted-sum
    dim3 gAttn(LSEQ, B);
    additive_attn_fused<<<gAttn, 256, 0, stream>>>(Qp, Kp, value, v_w, v_b,
                                                   (float*)d_out);
}